// SynapForgeHybridBlock_69011534512405
// MI455X (gfx1250) — compile-verified
//
#include <hip/hip_runtime.h>
#include <hip/hip_bf16.h>

// ---- problem constants (fixed by the reference) ----
#define BATCH 8
#define TLEN  2048
#define HDIM  256
#define BT    (BATCH * TLEN)      // 16384 rows
// Block = 8 waves; each wave computes a 64x16 output tile.
// Grid: 32 m-groups (of 8*64=512 rows) x 16 n-tiles = 512 blocks.

typedef __attribute__((ext_vector_type(2))) float v2f;
typedef __attribute__((ext_vector_type(8))) float v8f;

// -------------------------------------------------------------------------
// Kernel A: fused gate GEMMs with LDS-staged weights.
//   za = x @ W_a + b_a ; zu = x @ W_u + b_u
//   a  = sigmoid(za)   ; u  = tanh(zu)
//   store interleaved (a, (1-a)*u) as float2 for the scan kernel.
// Each block pins one 16-column N-tile: the needed strips of W_a/W_u
// (256x16 each, 32 KB total) are staged in LDS once and every wave reads
// its B fragments from LDS (ds_load), cutting global weight traffic ~32x.
// Each wave computes 64x16 of BOTH gates: 4 M-sub-tiles share each B
// fragment -> 8 x V_WMMA_F32_16X16X4_F32 per K-step (exact fp32; spikes
// are threshold-sensitive so we never downconvert).
// -------------------------------------------------------------------------
__global__ __launch_bounds__(256) void gates_gemm(
    const float* __restrict__ X,
    const float* __restrict__ Wa, const float* __restrict__ ba,
    const float* __restrict__ Wu, const float* __restrict__ bu,
    float* __restrict__ ab /* 2 floats per (row,h) */) {
  __shared__ float lwa[HDIM * 16];   // [k][n'] k-major, 16 KB
  __shared__ float lwu[HDIM * 16];   // [k][n']

  const int nt = blockIdx.x & 15;          // N tile (fixed per block)
  const int mg = blockIdx.x >> 4;          // m-group (0..31)
  const int lane = threadIdx.x & 31;
  const int wave = threadIdx.x >> 5;

  // ---- stage W_a / W_u column strips into LDS (coalesced float4) ----
  {
    const float4* ga = (const float4*)(Wa + nt * 16);
    const float4* gu = (const float4*)(Wu + nt * 16);
    float4* sa = (float4*)lwa;
    float4* su = (float4*)lwu;
    #pragma unroll
    for (int i = threadIdx.x; i < HDIM * 4; i += 256) {   // 1024 float4s
      const int k = i >> 2;
      const int q = i & 3;
      sa[i] = ga[k * (HDIM / 4) + q];     // W[k][nt*16 + q*4 .. +3]
      su[i] = gu[k * (HDIM / 4) + q];
    }
  }
  __syncthreads();

  const int hi   = lane >> 4;              // 0: lanes 0-15, 1: lanes 16-31
  const int l16  = lane & 15;
  const int n    = nt * 16 + l16;          // output column
  const int koff = hi << 1;                // A/B K sub-offset {0,2}
  const int m0   = (mg * 8 + wave) * 64;   // first row of this wave's tile

  v8f ca[4] = {v8f{}, v8f{}, v8f{}, v8f{}};
  v8f cu[4] = {v8f{}, v8f{}, v8f{}, v8f{}};

  const float* arow = X + (size_t)(m0 + l16) * HDIM + koff;
  #pragma unroll 2
  for (int k0 = 0; k0 < HDIM; k0 += 4) {
    const int k = k0 + koff;
    v2f bwa; bwa.x = lwa[k * 16 + l16]; bwa.y = lwa[(k + 1) * 16 + l16];
    v2f bwu; bwu.x = lwu[k * 16 + l16]; bwu.y = lwu[(k + 1) * 16 + l16];
    #pragma unroll
    for (int s = 0; s < 4; ++s) {
      v2f af = *(const v2f*)(arow + (size_t)(16 * s) * HDIM + k0);
      ca[s] = __builtin_amdgcn_wmma_f32_16x16x4_f32(false, af, false, bwa,
                                                    (short)0, ca[s], false, false);
      cu[s] = __builtin_amdgcn_wmma_f32_16x16x4_f32(false, af, false, bwu,
                                                    (short)0, cu[s], false, false);
    }
  }

  const float bia = ba[n];
  const float biu = bu[n];
  #pragma unroll
  for (int s = 0; s < 4; ++s) {
    const int rbase = m0 + 16 * s + (hi << 3);  // C/D: VGPR r -> row r (+8 upper)
    #pragma unroll
    for (int r = 0; r < 8; ++r) {
      const float za = ca[s][r] + bia;
      const float zu = cu[s][r] + biu;
      const float ga = 1.0f / (1.0f + __expf(-za));
      const float gu = tanhf(zu);
      v2f e; e.x = ga; e.y = (1.0f - ga) * gu;
      *(v2f*)(ab + 2 * ((size_t)(rbase + r) * HDIM + n)) = e;
    }
  }
}

// -------------------------------------------------------------------------
// Kernel B: fused liquid recurrence + PLIF. Both are serial in t (PLIF's
// nonlinear subtract-reset forbids a parallel scan), so fuse the cheap
// associative scan into the same sweep; per-step chain is ~4 dependent
// VALU ops, which is the latency floor for this workload. One lane per
// (b,h) sequence; 8B loads coalesced across h; prefetch + unroll-4 keeps
// the memory pipe ahead of the dependent chain.
// -------------------------------------------------------------------------
__global__ __launch_bounds__(256) void scan_plif(
    const float* __restrict__ ab, float* __restrict__ s_out,
    const float* __restrict__ tau_p, const float* __restrict__ thr_p) {
  const int idx = blockIdx.x * blockDim.x + threadIdx.x;  // 0..2047
  const int b = idx >> 8;          // / HDIM
  const int h = idx & (HDIM - 1);
  const float inv_tau = 1.0f / tau_p[0];
  const float thr = thr_p[0];
  const float decay = 1.0f - inv_tau;

  float hstate = 0.0f;
  float v = 0.0f;
  const size_t base = ((size_t)b * TLEN) * HDIM + h;

  #pragma unroll 4
  for (int t = 0; t < TLEN; ++t) {
    const size_t off = base + (size_t)t * HDIM;
    if (t + 16 < TLEN)  // run the memory pipe ahead of the serial chain
      __builtin_prefetch(ab + 2 * (base + (size_t)(t + 16) * HDIM), 0, 0);
    v2f e = *(const v2f*)(ab + 2 * off);
    hstate = e.x * hstate + e.y;          // h_t = a_t*h_{t-1} + (1-a_t)*u_t
    v = decay * v + inv_tau * hstate;     // PLIF charge (one FMA chain)
    const float sp = (v > thr) ? 1.0f : 0.0f;
    v -= sp * thr;                        // subtract reset
    s_out[off] = sp;
  }
}

// -------------------------------------------------------------------------
// Kernel C: synapse GEMM  out[r,o] = sum_h s[r,h] * W_syn[o,h]
// B[k][n] = W_syn[n][k]: the block's 16 W_syn rows (16 KB) are staged in
// LDS; B fragments become contiguous 8-byte ds loads. Each wave computes
// 64x16, sharing each B fragment across 4 M-sub-tiles.
// -------------------------------------------------------------------------
__global__ __launch_bounds__(256) void synapse_gemm(
    const float* __restrict__ S, const float* __restrict__ Wsyn,
    float* __restrict__ out) {
  __shared__ float lws[16 * HDIM];   // [n'][k] row-major, 16 KB

  const int nt = blockIdx.x & 15;
  const int mg = blockIdx.x >> 4;
  const int lane = threadIdx.x & 31;
  const int wave = threadIdx.x >> 5;

  // ---- stage 16 rows of W_syn into LDS ----
  {
    const float4* g = (const float4*)(Wsyn + (size_t)nt * 16 * HDIM);
    float4* sm = (float4*)lws;
    #pragma unroll
    for (int i = threadIdx.x; i < 16 * HDIM / 4; i += 256)  // 1024 float4s
      sm[i] = g[i];
  }
  __syncthreads();

  const int hi   = lane >> 4;
  const int l16  = lane & 15;
  const int n    = nt * 16 + l16;
  const int koff = hi << 1;
  const int m0   = (mg * 8 + wave) * 64;

  v8f c[4] = {v8f{}, v8f{}, v8f{}, v8f{}};

  const float* arow = S + (size_t)(m0 + l16) * HDIM + koff;
  const float* brow = lws + l16 * HDIM + koff;   // row n' of staged W_syn
  #pragma unroll 2
  for (int k0 = 0; k0 < HDIM; k0 += 4) {
    v2f bf = *(const v2f*)(brow + k0);           // contiguous 8B LDS load
    #pragma unroll
    for (int s = 0; s < 4; ++s) {
      v2f af = *(const v2f*)(arow + (size_t)(16 * s) * HDIM + k0);
      c[s] = __builtin_amdgcn_wmma_f32_16x16x4_f32(false, af, false, bf,
                                                   (short)0, c[s], false, false);
    }
  }

  #pragma unroll
  for (int s = 0; s < 4; ++s) {
    const int rbase = m0 + 16 * s + (hi << 3);
    #pragma unroll
    for (int r = 0; r < 8; ++r)
      out[(size_t)(rbase + r) * HDIM + n] = c[s][r];
  }
}

// -------------------------------------------------------------------------
extern "C" void kernel_launch(void* const* d_in, const int* in_sizes, int n_in,
                              void* d_out, int out_size, void* d_ws, size_t ws_size,
                              hipStream_t stream) {
  const float* x    = (const float*)d_in[0];  // (8, 2048, 256)
  const float* Wa   = (const float*)d_in[1];  // (256, 256)
  const float* ba   = (const float*)d_in[2];  // (256,)
  const float* Wu   = (const float*)d_in[3];  // (256, 256)
  const float* bu   = (const float*)d_in[4];  // (256,)
  const float* tau  = (const float*)d_in[5];  // scalar
  const float* thr  = (const float*)d_in[6];  // scalar
  const float* Wsyn = (const float*)d_in[7];  // (256, 256)
  float* out = (float*)d_out;                 // (8, 2048, 256)

  // workspace: [ab pairs: 2*BT*H floats = 32 MB][spikes: BT*H floats = 16 MB]
  float* ab = (float*)d_ws;
  float* s  = ab + 2 * (size_t)BT * HDIM;

  // 32 m-groups x 16 n-tiles = 512 blocks, 8 waves each (64x16 per wave)
  gates_gemm<<<512, 256, 0, stream>>>(x, Wa, ba, Wu, bu, ab);
  // 2048 sequences, one lane each
  scan_plif<<<8, 256, 0, stream>>>(ab, s, tau, thr);
  synapse_gemm<<<512, 256, 0, stream>>>(s, Wsyn, out);
}